// OrdLoss_82454782149225
// MI455X (gfx1250) — compile-verified
//
#include <hip/hip_runtime.h>

// Problem constants (fixed by the reference).
#define NB      2
#define CC      16
#define DDIM    32
#define HH      128
#define WW      128
#define DHW     (DDIM * HH * WW)      // 524288
#define MTOT    (NB * DHW)            // 1048576 voxels
#define QTOT    (MTOT / 4)            // 262144 quads of 4 voxels
#define BLOCKS  1024
#define THREADS 256                   // BLOCKS*THREADS == QTOT exactly

typedef __attribute__((ext_vector_type(2))) float v2f;
typedef __attribute__((ext_vector_type(8))) float v8f;

// One ordinal-loss term, computed unconditionally (validity applied once per
// voxel by the caller). Argument selected BEFORE the log so each element costs
// exactly one transcendental. Upper clamp (1e8) omitted: pred in [0,1) =>
// arg <= 1, so it can never fire; lower clamp keeps the log finite so the
// later multiply-by-0 for invalid voxels is safe (no NaN/Inf * 0).
__device__ __forceinline__ float ord_term(float p, int k, int t) {
  float arg = (k <= t) ? p : (1.0f - p);
  return __logf(fmaxf(arg, 1e-8f));
}

__global__ __launch_bounds__(THREADS) void ordloss_partial(
    const float* __restrict__ pred,
    const int*   __restrict__ target,   // only channel-0 slice is read (4MB not 64MB)
    const int*   __restrict__ mask,
    float* __restrict__ ws_s,           // [BLOCKS] partial log-sums
    float* __restrict__ ws_c)           // [BLOCKS] partial valid counts
{
  const int q = blockIdx.x * THREADS + threadIdx.x;   // quad id, exact cover
  float s = 0.0f, cnt = 0.0f;

  if (q < QTOT) {
    const int v  = q * 4;               // first voxel of the quad
    const int n  = v / DHW;
    const int sp = v - n * DHW;         // spatial offset, multiple of 4

    const int4 m4 = *(const int4*)(mask + v);
    const size_t base = (size_t)n * CC * DHW + sp;    // channel-0 offset
    const int4 t4 = *(const int4*)(target + base);

    const float va = (m4.x > 0) ? 1.0f : 0.0f;
    const float vb = (m4.y > 0) ? 1.0f : 0.0f;
    const float vc = (m4.z > 0) ? 1.0f : 0.0f;
    const float vd = (m4.w > 0) ? 1.0f : 0.0f;
    cnt = va + vb + vc + vd;

    // Per-voxel sums of 16 logs, validity folded in once at the end.
    float sx = 0.0f, sy = 0.0f, sz = 0.0f, sw = 0.0f;
#pragma unroll
    for (int k = 0; k < CC; ++k) {
      const float4 p4 = *(const float4*)(pred + base + (size_t)k * DHW);
      sx += ord_term(p4.x, k, t4.x);
      sy += ord_term(p4.y, k, t4.y);
      sz += ord_term(p4.z, k, t4.z);
      sw += ord_term(p4.w, k, t4.w);
    }
    s = fmaf(va, sx, fmaf(vb, sy, fmaf(vc, sz, vd * sw)));
  }

  // Deterministic block reduction (fixed tree order).
  __shared__ float red_s[THREADS];
  __shared__ float red_c[THREADS];
  red_s[threadIdx.x] = s;
  red_c[threadIdx.x] = cnt;
  __syncthreads();
#pragma unroll
  for (int off = THREADS / 2; off > 0; off >>= 1) {
    if (threadIdx.x < off) {
      red_s[threadIdx.x] += red_s[threadIdx.x + off];
      red_c[threadIdx.x] += red_c[threadIdx.x + off];
    }
    __syncthreads();
  }
  if (threadIdx.x == 0) {
    ws_s[blockIdx.x] = red_s[0];
    ws_c[blockIdx.x] = red_c[0];
  }
}

// Final reduction: one wave32. The 1024 partials are summed on the matrix
// unit: D = A(16x4) * ones(4x16) + C gives row sums chained through the f32
// accumulator -> 16 v_wmma_f32_16x16x4_f32 issues sum all partials.
// Column N=0 of the accumulator lives in lane 0 (M=0..7) and lane 16
// (M=8..15), so total = lanesum(lane0) + lanesum(lane16).
__global__ __launch_bounds__(32) void ordloss_final(
    const float* __restrict__ ws_s,
    const float* __restrict__ ws_c,
    float* __restrict__ out)
{
  const int lane = threadIdx.x;        // 32 lanes, uniform control flow (EXEC all 1s)

  v8f acc = {};
  v2f ones; ones.x = 1.0f; ones.y = 1.0f;

#pragma unroll
  for (int it = 0; it < BLOCKS / 64; ++it) {   // 16 iterations x 64 values
    v2f a;
    a.x = ws_s[it * 64 + lane];
    a.y = ws_s[it * 64 + 32 + lane];
    // (neg_a, A, neg_b, B, c_mod, C, reuse_a, reuse_b)
    acc = __builtin_amdgcn_wmma_f32_16x16x4_f32(
        false, a, false, ones, (short)0, acc, false, false);
  }

  float lanesum = acc[0] + acc[1] + acc[2] + acc[3] +
                  acc[4] + acc[5] + acc[6] + acc[7];
  const float s_total = lanesum + __shfl(lanesum, lane ^ 16, 32); // valid on lane 0

  // Valid-voxel count: fixed-trip uniform loop (no exec masking), then a
  // deterministic wave butterfly.
  float c = 0.0f;
#pragma unroll
  for (int it = 0; it < BLOCKS / 32; ++it) c += ws_c[it * 32 + lane];
#pragma unroll
  for (int off = 16; off > 0; off >>= 1) c += __shfl_xor(c, off, 32);

  if (lane == 0) out[0] = s_total / (-c);
}

extern "C" void kernel_launch(void* const* d_in, const int* in_sizes, int n_in,
                              void* d_out, int out_size, void* d_ws, size_t ws_size,
                              hipStream_t stream) {
  (void)in_sizes; (void)n_in; (void)out_size; (void)ws_size;
  const float* pred   = (const float*)d_in[0];
  const int*   target = (const int*)d_in[1];
  const int*   mask   = (const int*)d_in[2];
  float* out = (float*)d_out;

  float* ws_s = (float*)d_ws;          // BLOCKS floats
  float* ws_c = ws_s + BLOCKS;         // BLOCKS floats (8 KB total scratch)

  ordloss_partial<<<BLOCKS, THREADS, 0, stream>>>(pred, target, mask, ws_s, ws_c);
  ordloss_final<<<1, 32, 0, stream>>>(ws_s, ws_c, out);
}